// RPN_89756226552207
// MI455X (gfx1250) — compile-verified
//
#include <hip/hip_runtime.h>
#include <math.h>
#include <stdint.h>

// Anchor counts per FPN level (3 aspect ratios each)
#define N3 28812   // 98*98*3
#define N4 7203    // 49*49*3
#define N5 1875    // 25*25*3
#define NANCH 37890
#define NGT 64

#define OFF_PROP 0
#define OFF_MATCH (NANCH * 4)   // 151560
#define OFF_GTD  (NANCH * 9)    // 341010

__launch_bounds__(256)
__global__ void rpn_fused_kernel(const float* __restrict__ deltas,
                                 const float* __restrict__ gt,   // [64,5]
                                 float* __restrict__ out)        // [NANCH*13]
{
    // GT table staged in LDS: 64 boxes * 5 f32 = 1280 bytes = 80 x b128.
    __shared__ float sgt[NGT * 5];

    const int tid = threadIdx.x;

    // ---- CDNA5 async global->LDS copy of the GT table (ASYNCcnt path) ----
    if (tid < 80) {
        uint32_t lds_addr = (uint32_t)(uintptr_t)(&sgt[0]) + (uint32_t)(tid * 16);
        uint32_t goff     = (uint32_t)(tid * 16);
        uint64_t gbase    = (uint64_t)(uintptr_t)gt;
        // GVS form: vdst = LDS byte address, vaddr = 32-bit offset, saddr = 64-bit base
        asm volatile("global_load_async_to_lds_b128 %0, %1, %2"
                     :: "v"(lds_addr), "v"(goff), "s"(gbase)
                     : "memory");
    }
#if __has_builtin(__builtin_amdgcn_s_wait_asynccnt)
    __builtin_amdgcn_s_wait_asynccnt(0);
#else
    asm volatile("s_wait_asynccnt 0" ::: "memory");
#endif
    __syncthreads();

    const int i = blockIdx.x * blockDim.x + tid;
    if (i >= NANCH) return;

    // ---- Decode anchor analytically from index (exact f32 reproduction) ----
    int j, hwgrid;
    float s;
    if (i < N3)            { j = i;             hwgrid = 98; s = 8.0f;  }
    else if (i < N3 + N4)  { j = i - N3;        hwgrid = 49; s = 16.0f; }
    else                   { j = i - (N3 + N4); hwgrid = 25; s = 32.0f; }
    const int loc = j / 3;
    const int ar  = j - loc * 3;
    const int row = loc / hwgrid;
    const int col = loc - row * hwgrid;
    const float xc = ((float)col + 0.5f) * s;   // exact: s is a power of two
    const float yc = ((float)row + 0.5f) * s;

    // half-extents: ar=0.5 -> (2sqrt2*s, sqrt2*s); ar=1 -> (2s,2s); ar=2 -> (sqrt2*s, 2sqrt2*s)
    // f32(2sqrt2)*s == f32(double(2sqrt2)*s) because s = 2^k (exact exponent scaling).
    const float A = 2.8284271247461903f;   // 2*sqrt(2)
    const float B = 1.4142135623730951f;   // sqrt(2)
    float hwid, hhei;
    if (ar == 0)      { hwid = A * s; hhei = B * s; }
    else if (ar == 1) { hwid = 2.0f * s; hhei = 2.0f * s; }
    else              { hwid = B * s; hhei = A * s; }

    const float ax1 = xc - hwid, ay1 = yc - hhei;
    const float ax2 = xc + hwid, ay2 = yc + hhei;
    const float wa = ax2 - ax1, ha = ay2 - ay1;
    const float xa = ax1 + wa * 0.5f, ya = ay1 + ha * 0.5f;

    // ---- proposals = apply_deltas(deltas, anchors) ----
    const float4 dlt = ((const float4*)deltas)[i];   // global_load_b128
    {
        const float SCALE_CLAMP = 3.3322045101752038f;  // log(224/8)
        const float dw = fminf(dlt.z, SCALE_CLAMP);
        const float dh = fminf(dlt.w, SCALE_CLAMP);
        const float xp = dlt.x * wa + xa;
        const float yp = dlt.y * ha + ya;
        const float wp = expf(dw) * wa;
        const float hp = expf(dh) * ha;
        float* p = out + OFF_PROP + 4 * i;
        p[0] = xp - wp * 0.5f;
        p[1] = yp - hp * 0.5f;
        p[2] = xp + wp * 0.5f;
        p[3] = yp + hp * 0.5f;
    }

    // ---- IoU vs 64 GT boxes, argmax (first-max-wins like jnp.argmax) ----
    const float areaA = wa * ha;
    float best = -1.0f;
    int bi = 0;
#pragma unroll 4
    for (int g = 0; g < NGT; ++g) {
        const float gx1 = sgt[g * 5 + 0];
        const float gy1 = sgt[g * 5 + 1];
        const float gx2 = sgt[g * 5 + 2];
        const float gy2 = sgt[g * 5 + 3];
        const float ltx = fmaxf(ax1, gx1);
        const float lty = fmaxf(ay1, gy1);
        const float rbx = fminf(ax2, gx2);
        const float rby = fminf(ay2, gy2);
        const float iw = fmaxf(rbx - ltx, 0.0f);
        const float ih = fmaxf(rby - lty, 0.0f);
        const float inter = iw * ih;
        const float uni = areaA + (gx2 - gx1) * (gy2 - gy1) - inter;
        const float iou = inter / uni;
        if (iou > best) { best = iou; bi = g; }
    }

    // ---- matched_gt with thresholding (replicates the two jnp.where passes) ----
    float m0 = sgt[bi * 5 + 0];
    float m1 = sgt[bi * 5 + 1];
    float m2 = sgt[bi * 5 + 2];
    float m3 = sgt[bi * 5 + 3];
    float m4 = sgt[bi * 5 + 4];
    if (best <= 0.3f)      { m0 = m1 = m2 = m3 = m4 = -1.0f;   }
    else if (best < 0.6f)  { m0 = m1 = m2 = m3 = m4 = -1.0e8f; }
    {
        float* m = out + OFF_MATCH + 5 * i;
        m[0] = m0; m[1] = m1; m[2] = m2; m[3] = m3; m[4] = m4;
    }

    // ---- gt_deltas ----
    float o0, o1, o2, o3;
    if (m0 < 0.0f) {
        o0 = o1 = o2 = o3 = -1.0e8f;
    } else {
        const float wg = fmaxf(m2 - m0, 1.0f);
        const float hg = fmaxf(m3 - m1, 1.0f);
        const float xg = m0 + wg * 0.5f;
        const float yg = m1 + hg * 0.5f;
        o0 = (xg - xa) / wa;
        o1 = (yg - ya) / ha;
        o2 = logf(wg / wa);
        o3 = logf(hg / ha);
    }
    {
        float* d = out + OFF_GTD + 4 * i;
        d[0] = o0; d[1] = o1; d[2] = o2; d[3] = o3;
    }
}

extern "C" void kernel_launch(void* const* d_in, const int* in_sizes, int n_in,
                              void* d_out, int out_size, void* d_ws, size_t ws_size,
                              hipStream_t stream) {
    (void)in_sizes; (void)n_in; (void)out_size; (void)d_ws; (void)ws_size;
    // Inputs: 0=locations_p3, 1=locations_p4, 2=locations_p5 (unused; recomputed
    // analytically), 3=deltas [37890,4] f32, 4=gt_boxes [64,5] f32.
    const float* deltas = (const float*)d_in[3];
    const float* gt     = (const float*)d_in[4];
    float* out          = (float*)d_out;

    const int threads = 256;                       // 8 wave32s per block
    const int blocks  = (NANCH + threads - 1) / threads;  // 149
    rpn_fused_kernel<<<blocks, threads, 0, stream>>>(deltas, gt, out);
}